// EC_Layer_23613730193431
// MI455X (gfx1250) — compile-verified
//
#include <hip/hip_runtime.h>

typedef __attribute__((ext_vector_type(2))) float v2f;
typedef __attribute__((ext_vector_type(8))) float v8f;

#define NSLICE  96          // 32 * 3
#define NPTS    8192
#define TBINS   256
#define DSTRIDE 272         // 257 used, padded
#define CHUNKS  4
#define CPTS    (NPTS / CHUNKS)

// ---------------------------------------------------------------- zero ws
__global__ void ec_zero_ws(int* __restrict__ ws, int n) {
    int i = blockIdx.x * blockDim.x + threadIdx.x;
    if (i < n) ws[i] = 0;
}

// ---------------------------------------------------------------- scatter
// One block per (slice, pd, point-chunk). Builds a signed difference
// histogram in LDS, then merges non-zero bins into the per-slice global
// diff array with global atomics (order-independent -> deterministic).
__global__ __launch_bounds__(256) void ec_scatter(const float2* __restrict__ pd0,
                                                  const float2* __restrict__ pd1,
                                                  int* __restrict__ ws) {
    __shared__ int ldiff[DSTRIDE];
    const int t = threadIdx.x;
    for (int i = t; i < DSTRIDE; i += 256) ldiff[i] = 0;
    __syncthreads();

    const int bx    = blockIdx.x;
    const int slice = bx >> 3;
    const int sub   = bx & 7;
    const int which = sub >> 2;        // 0 -> pd0 (+1), 1 -> pd1 (-1)
    const int chunk = sub & 3;
    const float2* __restrict__ src =
        (which ? pd1 : pd0) + (size_t)slice * NPTS + (size_t)chunk * CPTS;
    const int s = which ? -1 : 1;

    const float inv = 1.0f / 255.0f;
#pragma unroll
    for (int i = 0; i < CPTS / 256; ++i) {
        float2 p = src[t + i * 256];
        float b = p.x, d = p.y;
        // lo = smallest i with birth <  i/255
        // hi = largest  i with death >= i/255
        int lo = (int)floorf(b * 255.0f) + 1;
        int hi = (int)floorf(d * 255.0f);
        // 1-ULP fixups against t_i = i * (1/255)
        if (!(b < (float)lo * inv))            lo++;
        else if (b < (float)(lo - 1) * inv)    lo--;
        if (!(d >= (float)hi * inv))           hi--;
        else if (d >= (float)(hi + 1) * inv)   hi++;
        lo = max(lo, 0);
        hi = min(hi, 255);
        if (lo <= hi) {
            atomicAdd(&ldiff[lo], s);
            atomicAdd(&ldiff[hi + 1], -s);
        }
    }
    __syncthreads();

    int* __restrict__ g = ws + slice * DSTRIDE;
    for (int i = t; i < 257; i += 256) {
        int v = ldiff[i];
        if (v) atomicAdd(&g[i], v);
    }
}

// ---------------------------------------------------------------- WMMA scan
// betti = S + Ls*G where S = Dm*U, G = Dm*J (Dm = 16x16 reshape of diff).
__device__ __forceinline__ v8f wmma4(v2f a, v2f b, v8f c) {
    // V_WMMA_F32_16X16X4_F32 : (neg_a, A, neg_b, B, c_mod, C, reuse_a, reuse_b)
    return __builtin_amdgcn_wmma_f32_16x16x4_f32(
        false, a, false, b, (short)0, c, false, false);
}

__global__ __launch_bounds__(32) void ec_scan_wmma(const int* __restrict__ ws,
                                                   float* __restrict__ out) {
    __shared__ float Gs[256];
    const int slice = blockIdx.x;
    const int* __restrict__ diff = ws + slice * DSTRIDE;
    const int lane = threadIdx.x;        // single wave, EXEC all ones
    const int half = lane >> 4;          // A/B frag: K = v + 2*half (+4*kb)
    const int n    = lane & 15;          // A frag: M = n ; B frag: N = n

    // A fragments of Dm: Dm[M][K] = diff[16*M + K]
    v2f aD[4];
#pragma unroll
    for (int kb = 0; kb < 4; ++kb) {
        aD[kb].x = (float)diff[16 * n + 4 * kb + 0 + 2 * half];
        aD[kb].y = (float)diff[16 * n + 4 * kb + 1 + 2 * half];
    }

    v8f S = {0.f, 0.f, 0.f, 0.f, 0.f, 0.f, 0.f, 0.f};
    v8f G = {0.f, 0.f, 0.f, 0.f, 0.f, 0.f, 0.f, 0.f};
#pragma unroll
    for (int kb = 0; kb < 4; ++kb) {
        const int k0 = 4 * kb + 2 * half;
        const int k1 = k0 + 1;
        v2f bU, bJ;
        bU.x = (k0 <= n) ? 1.0f : 0.0f;   // U upper-triangular (incl.)
        bU.y = (k1 <= n) ? 1.0f : 0.0f;
        bJ.x = 1.0f; bJ.y = 1.0f;         // J all ones
        S = wmma4(aD[kb], bU, S);         // S += Dm[:,kb] * U[kb,:]
        G = wmma4(aD[kb], bJ, G);         // G += Dm[:,kb] * J[kb,:]
    }

    // accumulator -> B-fragment layout via LDS (G[M][N] row-major)
#pragma unroll
    for (int v = 0; v < 8; ++v) Gs[16 * (v + 8 * half) + n] = G[v];
    __syncthreads();                      // single-wave WG: cheap

#pragma unroll
    for (int kb = 0; kb < 4; ++kb) {
        const int k0 = 4 * kb + 2 * half;
        const int k1 = k0 + 1;
        v2f aL, bG;
        aL.x = (k0 < n) ? 1.0f : 0.0f;    // Ls strict lower-triangular
        aL.y = (k1 < n) ? 1.0f : 0.0f;
        bG.x = Gs[16 * k0 + n];
        bG.y = Gs[16 * k1 + n];
        S = wmma4(aL, bG, S);             // S += Ls[:,kb] * G[kb,:]
    }

    float* __restrict__ o = out + (size_t)slice * TBINS;
#pragma unroll
    for (int v = 0; v < 8; ++v) o[16 * (v + 8 * half) + n] = S[v];
}

// ---------------------------------------------------------------- launch
extern "C" void kernel_launch(void* const* d_in, const int* in_sizes, int n_in,
                              void* d_out, int out_size, void* d_ws, size_t ws_size,
                              hipStream_t stream) {
    const float2* pd0 = (const float2*)d_in[0];
    const float2* pd1 = (const float2*)d_in[1];
    int*   ws  = (int*)d_ws;      // 96 * 272 ints = 104448 B
    float* out = (float*)d_out;

    const int nz = NSLICE * DSTRIDE;
    ec_zero_ws<<<(nz + 255) / 256, 256, 0, stream>>>(ws, nz);
    ec_scatter<<<NSLICE * 2 * CHUNKS, 256, 0, stream>>>(pd0, pd1, ws);
    ec_scan_wmma<<<NSLICE, 32, 0, stream>>>(ws, out);
}